// DropStripes_44710609551576
// MI455X (gfx1250) — compile-verified
//
#include <hip/hip_runtime.h>

// DropStripes (SpecAugment along time axis): out = x with time rows
// [bgn, bgn+distance) zeroed per sample, 2 stripes per sample.
// x: [B=64, C=1, T=4096, F=256] f32. Pure streaming op: 512 MiB traffic,
// ~22 us floor at 23.3 TB/s HBM.
//
// Layout facts exploited:
//   - one row = 256 f32 = 64 float4; one wave32 covers half a row -> t is
//     wave-uniform -> the drop test becomes a scalar-exec branch that skips
//     the global load for dropped rows.
//   - one 256-thread block = 4 rows of ONE sample (1024 blocks per sample)
//     -> b = blockIdx.x >> 10 is scalar -> stripe params lower to s_load
//     (KMcnt path) instead of per-lane VMEM loads.

typedef float v4f __attribute__((ext_vector_type(4)));

static constexpr int kB = 64;
static constexpr int kT = 4096;                    // 2^12
static constexpr int kF = 256;                     // 64 float4 per row
static constexpr int kVecPerRow = kF / 4;          // 64
static constexpr long long kNVec = (long long)kB * kT * kVecPerRow; // 16,777,216
static constexpr int kTPB = 256;                   // 8 wave32; block = 4 rows
static constexpr int kRowsPerBlock = kTPB / kVecPerRow;             // 4
static constexpr int kBlocksPerSample = kT / kRowsPerBlock;         // 1024

__global__ __launch_bounds__(kTPB) void drop_stripes_kernel(
    const v4f* __restrict__ x,
    const int* __restrict__ bgn,      // [B, 2]
    const int* __restrict__ dist,     // [B, 2]
    v4f* __restrict__ out)
{
    const unsigned blk = blockIdx.x;
    const unsigned tid = threadIdx.x;

    // Scalar sample index -> stripe params become s_load (scalar constant path).
    const int b  = (int)(blk >> 10);                 // blk / kBlocksPerSample
    const int b0 = bgn[b * 2 + 0];
    const int d0 = dist[b * 2 + 0];
    const int b1 = bgn[b * 2 + 1];
    const int d1 = dist[b * 2 + 1];

    // Wave-uniform time index (threadIdx>>6 is constant across a wave32).
    const int t = (int)(((blk * kRowsPerBlock) + (tid >> 6)) & (kT - 1));

    // t in [b0, b0+d0) <=> (unsigned)(t-b0) < (unsigned)d0  (d < 64, no wrap).
    const bool drop = ((unsigned)(t - b0) < (unsigned)d0) |
                      ((unsigned)(t - b1) < (unsigned)d1);

    const long long i = (long long)blk * kTPB + tid;  // float4 index

    v4f v;
    if (drop) {
        // Wave-uniform branch: dropped rows skip the load, stream zeros.
        v = (v4f){0.0f, 0.0f, 0.0f, 0.0f};
    } else {
        // Single-touch streaming read: non-temporal (th:TH_LOAD_NT).
        v = __builtin_nontemporal_load(&x[i]);
    }
    // Single-touch streaming write: non-temporal (th:TH_STORE_NT).
    __builtin_nontemporal_store(v, &out[i]);
}

extern "C" void kernel_launch(void* const* d_in, const int* in_sizes, int n_in,
                              void* d_out, int out_size, void* d_ws, size_t ws_size,
                              hipStream_t stream) {
    (void)in_sizes; (void)n_in; (void)d_ws; (void)ws_size; (void)out_size;

    const v4f* x    = (const v4f*)d_in[0];   // float32 [B,C,T,F]
    const int* bgn  = (const int*)d_in[1];   // int32   [B,2]
    const int* dist = (const int*)d_in[2];   // int32   [B,2]
    v4f* out        = (v4f*)d_out;           // float32 [B,C,T,F]

    const long long nblocks = kNVec / kTPB;  // 65,536
    drop_stripes_kernel<<<dim3((unsigned)nblocks), dim3(kTPB), 0, stream>>>(
        x, bgn, dist, out);
}